// UpsamplingDepthToSpace_34230889349710
// MI455X (gfx1250) — compile-verified
//
#include <hip/hip_runtime.h>

// Problem constants (from reference): x[32,256,64,64] f32, r=2 -> out[32,64,128,128] f32
#define B_   32
#define C_   256
#define H_   64
#define W_   64
#define UPC_ 64     // C / r^2
#define OH_  128    // H * r
#define OW_  128    // W * r
#define CH_STRIDE_ (H_ * W_)          // 4096 floats per channel
#define IJ_STRIDE_ (64 * CH_STRIDE_)  // +64 channels per (i,j) group = 262144 floats

// ---- CDNA5 async LDS data-mover availability probe (compile-time) ----
#if defined(__has_builtin)
#if __has_builtin(__builtin_amdgcn_global_load_async_to_lds_b32) && \
    __has_builtin(__builtin_amdgcn_global_store_async_from_lds_b128)
#define USE_ASYNC 1
#endif
#endif
#ifndef USE_ASYNC
#define USE_ASYNC 0
#endif

#if USE_ASYNC
// Payload-typed pointers per the builtin signatures revealed by the compiler:
//   b32  -> int*           (AS1 global, AS3 lds)
//   b128 -> v4i*           (AS1 global, AS3 lds)
typedef int v4i __attribute__((vector_size(16)));
typedef __attribute__((address_space(1))) int  g_i32_t;
typedef __attribute__((address_space(3))) int  l_i32_t;
typedef __attribute__((address_space(1))) v4i  g_v4i_t;
typedef __attribute__((address_space(3))) v4i  l_v4i_t;

__device__ __forceinline__ void async_load_f32_to_lds(const float* g, float* l) {
  __builtin_amdgcn_global_load_async_to_lds_b32(
      (g_i32_t*)(g), (l_i32_t*)(l), /*offset=*/0, /*cpol=*/0);
}
__device__ __forceinline__ void async_store_b128_from_lds(float* g, const float* l) {
  __builtin_amdgcn_global_store_async_from_lds_b128(
      (g_v4i_t*)(g), (l_v4i_t*)(l), /*offset=*/0, /*cpol=*/0);
}
__device__ __forceinline__ void wait_asynccnt0() {
#if defined(__has_builtin) && __has_builtin(__builtin_amdgcn_s_wait_asynccnt)
  __builtin_amdgcn_s_wait_asynccnt(0);
#else
  asm volatile("s_wait_asynccnt 0x0" ::: "memory");
#endif
}

// One wave per (b,u,h) tile: 4 input rows (1KiB) async-scattered into a
// pre-permuted per-wave LDS tile, then 2 output rows drained with dense
// B128 async stores. The DCR permutation lives entirely in the addressing.
__global__ void __launch_bounds__(256) d2s_async(const float* __restrict__ x,
                                                 float* __restrict__ out) {
  __shared__ float tile[8 * 2 * OW_];      // 8 waves * 256 floats (1 KiB/wave)
  const int lane = threadIdx.x & 31;
  const int wave = threadIdx.x >> 5;
  const int t  = blockIdx.x * 8 + wave;    // tile id = (b*UPC + u)*H + h
  const int h  = t & (H_ - 1);
  const int bu = t >> 6;                   // b*UPC + u
  const int u  = bu & (UPC_ - 1);
  const int b  = bu >> 6;

  const float* xbase = x + (size_t)(b * C_ + u) * CH_STRIDE_ + (size_t)h * W_;
  float* lt = &tile[wave * (2 * OW_)];

  // Scatter-load: channel group ij = 2i+j, two 32-lane segments per 64-float row.
  // LDS position of x[.., (2i+j)*64+u, h, w] is  i*OW + 2w + j  (pre-permuted).
#pragma unroll
  for (int ij = 0; ij < 4; ++ij) {
    const int i = ij >> 1, j = ij & 1;
#pragma unroll
    for (int seg = 0; seg < 2; ++seg) {
      const int w = seg * 32 + lane;
      async_load_f32_to_lds(xbase + (size_t)ij * IJ_STRIDE_ + w,
                            lt + i * OW_ + 2 * w + j);
    }
  }

  wait_asynccnt0();

  // Dense drain: 2 output rows x 128 floats; 16B per lane = 512B per instruction.
  const int obase = (bu * OH_ + 2 * h) * OW_;
#pragma unroll
  for (int i = 0; i < 2; ++i) {
    async_store_b128_from_lds(out + obase + i * OW_ + 4 * lane,
                              lt + i * OW_ + 4 * lane);
  }
  // ASYNCcnt drains at s_endpgm (implicit wait-idle).
}
#endif // USE_ASYNC

// Fallback / reference-grade direct path: one thread per output float4.
// Two coalesced float2 loads (256B/wave each) + one float4 store (512B/wave).
__global__ void __launch_bounds__(256) d2s_direct(const float* __restrict__ x,
                                                  float* __restrict__ out) {
  const int tid = blockIdx.x * blockDim.x + threadIdx.x;  // output float4 id
  const int ow4 = tid & 31;                // 32 float4 per output row
  const int oh  = (tid >> 5) & (OH_ - 1);
  const int u   = (tid >> 12) & (UPC_ - 1);
  const int b   = tid >> 18;

  const int h = oh >> 1, i = oh & 1;
  const int c0 = i * 128 + u;              // (2i+0)*64 + u ; c1 = c0 + 64
  const size_t base = (size_t)(b * C_ + c0) * CH_STRIDE_ + (size_t)h * W_ + 2 * ow4;

  const float2 a  = *reinterpret_cast<const float2*>(x + base);
  const float2 bb = *reinterpret_cast<const float2*>(x + base + (size_t)IJ_STRIDE_);
  float4 o;
  o.x = a.x;  o.y = bb.x;  o.z = a.y;  o.w = bb.y;   // interleave j=0 / j=1
  reinterpret_cast<float4*>(out)[tid] = o;
}

extern "C" void kernel_launch(void* const* d_in, const int* in_sizes, int n_in,
                              void* d_out, int out_size, void* d_ws, size_t ws_size,
                              hipStream_t stream) {
  (void)in_sizes; (void)n_in; (void)out_size; (void)d_ws; (void)ws_size;
  const float* x = (const float*)d_in[0];
  float* out = (float*)d_out;
#if USE_ASYNC
  const int tiles = B_ * UPC_ * H_;                 // 131072 wave-tiles
  d2s_async<<<tiles / 8, 256, 0, stream>>>(x, out); // 8 waves per block
#else
  const int n4 = B_ * UPC_ * OH_ * OW_ / 4;         // 8388608 float4s
  d2s_direct<<<n4 / 256, 256, 0, stream>>>(x, out);
#endif
}